// GNNExtrapolation_70626442215851
// MI455X (gfx1250) — compile-verified
//
#include <hip/hip_runtime.h>
#include <math.h>

// Problem constants (from reference)
#define B_     4
#define T_IN_  12
#define T_OUT_ 24
#define N_     5000
#define C_     32
#define H_     8
#define K1_    17
#define TH_    (T_IN_ * H_)      // 96
#define OT_    (T_OUT_ - T_IN_)  // 12
#define OTP_   16                // padded N-columns for WMMA (zeros in 12..15)
#define KSTEPS (TH_ / 4)         // 24
#define BETA_  0.8f

typedef __attribute__((ext_vector_type(2))) float v2f;
typedef __attribute__((ext_vector_type(8))) float v8f;

// ---------------------------------------------------------------------------
// Kernel 1: sigma reduction.  ws[0] = 1/sigma^2 = 4 / max(d)^2
// ---------------------------------------------------------------------------
__global__ __launch_bounds__(256) void gnn_sigma_kernel(
    const float* __restrict__ nd, float* __restrict__ ws) {
  __shared__ float red[256];
  float m = 0.0f;
  for (int i = threadIdx.x; i < N_ * K1_; i += 256) m = fmaxf(m, nd[i]);
  red[threadIdx.x] = m;
  __syncthreads();
  for (int s = 128; s > 0; s >>= 1) {
    if (threadIdx.x < s) red[threadIdx.x] = fmaxf(red[threadIdx.x], red[threadIdx.x + s]);
    __syncthreads();
  }
  if (threadIdx.x == 0) {
    float mx = red[0];
    float sig = mx * 0.5f;          // SIGMA_RATIO = 2
    ws[0] = 1.0f / (sig * sig);
  }
}

// ---------------------------------------------------------------------------
// Kernel 2: fully fused gather-aggregate -> shrink matmul (WMMA) -> swish.
// One block (4 wave32) per node n; wave = batch b; lane = channel c.
// ---------------------------------------------------------------------------
__global__ __launch_bounds__(128) void gnn_fused_kernel(
    const float* __restrict__ x,       // (B, T_IN, N, C)
    const int*   __restrict__ nn,      // (N, K1)
    const float* __restrict__ nd,      // (N, K1)
    const float* __restrict__ Wm,      // (TH, OT) row-major
    const float* __restrict__ bias,    // (OT,)
    const float* __restrict__ ws,      // [0] = 1/sigma^2
    float*       __restrict__ out) {   // (B, T_OUT, N, C)
  __shared__ float w_lds[K1_ * H_];          // 136 GCN weights for this node
  __shared__ int   nn_lds[K1_];
  __shared__ float Wm_lds[TH_ * OTP_];       // 96x16, cols 12..15 zero-padded
  __shared__ float bias_lds[OTP_];           // padded: no divergent reads
  __shared__ float feat[B_][C_][TH_ + 1];    // pad 97: gcd(97,64)=1 -> no bank conflicts

  const int n    = blockIdx.x;
  const int tid  = threadIdx.x;
  const int lane = tid & 31;                 // channel c
  const int b    = tid >> 5;                 // batch  (4 waves)
  const float inv_s2 = ws[0];

  // ---- Stage 0: per-node GCN weights + zero-padded shrink weights in LDS --
  for (int i = tid; i < K1_ * H_; i += 128) {
    const int j = i / H_, h = i % H_;
    const float d    = nd[n * K1_ + j];
    const int   node = nn[n * K1_ + j];
    const float lam  = (float)(h + 1) * (1.0f / (float)H_);
    float wv = __expf(-d * d * lam * inv_s2);
    if (node < 0 || wv < 1e-5f) wv = 0.0f;   // missing neighbor / threshold
    w_lds[i] = wv;
  }
  if (tid < K1_) {
    const int node = nn[n * K1_ + tid];
    nn_lds[tid] = (node < 0) ? 0 : node;     // clamp; weight already zeroed
  }
  for (int i = tid; i < TH_ * OTP_; i += 128) {
    const int k = i >> 4, o = i & (OTP_ - 1);
    Wm_lds[i] = (o < OT_) ? Wm[k * OT_ + o] : 0.0f;
  }
  if (tid < OTP_) bias_lds[tid] = (tid < OT_) ? bias[tid] : 0.0f;
  __syncthreads();

  // ---- Stage 1: gather-weighted aggregation (coalesced 128B row loads) ----
  float acc[T_IN_][H_];
#pragma unroll
  for (int t = 0; t < T_IN_; ++t)
#pragma unroll
    for (int h = 0; h < H_; ++h) acc[t][h] = 0.0f;

  for (int j = 0; j < K1_; ++j) {
    const int m = nn_lds[j];
    float wj[H_];
#pragma unroll
    for (int h = 0; h < H_; ++h) wj[h] = w_lds[j * H_ + h];
    const float* xp = x + (((size_t)b * T_IN_) * N_ + m) * C_ + lane;
#pragma unroll
    for (int t = 0; t < T_IN_; ++t) {
      const float xv = xp[(size_t)t * N_ * C_];
#pragma unroll
      for (int h = 0; h < H_; ++h) acc[t][h] = fmaf(xv, wj[h], acc[t][h]);
    }
  }

  // ---- passthrough copy: out[:, 0:T_IN] = x -------------------------------
#pragma unroll
  for (int t = 0; t < T_IN_; ++t) {
    out[(((size_t)b * T_OUT_ + t) * N_ + n) * C_ + lane] =
        x[(((size_t)b * T_IN_ + t) * N_ + n) * C_ + lane];
  }

  // ---- Stage 2: feat (C x 96) per batch into LDS --------------------------
#pragma unroll
  for (int t = 0; t < T_IN_; ++t)
#pragma unroll
    for (int h = 0; h < H_; ++h) feat[b][lane][t * H_ + h] = acc[t][h];
  __syncthreads();

  // ---- Stage 3: shrink matmul on the matrix pipe --------------------------
  // D(16x16) = A(16x4) x B(4x16) + C, f32 (V_WMMA_F32_16X16X4_F32).
  // f32 A layout: lanes 0-15 hold K=k0,k0+1 (v0,v1); lanes 16-31 hold K=k0+2,k0+3.
  // f32 B layout: same K split across lane halves, N = lane&15.
  const int nidx  = lane & 15;          // N column (output time index o)
  const int khalf = (lane >> 4) * 2;    // 0 for lanes 0-15, 2 for lanes 16-31

  // Hoist all B fragments (shared by both M tiles) into registers.
  v2f bfrag[KSTEPS];
#pragma unroll
  for (int kk = 0; kk < KSTEPS; ++kk) {
    const int k0 = kk * 4 + khalf;
    bfrag[kk].x = Wm_lds[(k0)     * OTP_ + nidx];   // zeros for nidx >= 12
    bfrag[kk].y = Wm_lds[(k0 + 1) * OTP_ + nidx];
  }
  const float bv = bias_lds[nidx];

#pragma unroll
  for (int mt = 0; mt < 2; ++mt) {      // C=32 rows -> two 16-row M tiles
    v8f c8 = {0.f, 0.f, 0.f, 0.f, 0.f, 0.f, 0.f, 0.f};
    const int mrow = mt * 16 + nidx;    // A row = channel index
#pragma unroll
    for (int kk = 0; kk < KSTEPS; ++kk) {   // 24 K-steps of 4
      const int k0 = kk * 4 + khalf;
      v2f a;
      a.x = feat[b][mrow][k0];
      a.y = feat[b][mrow][k0 + 1];
      c8 = __builtin_amdgcn_wmma_f32_16x16x4_f32(
          /*neg_a=*/false, a, /*neg_b=*/false, bfrag[kk],
          /*c_mod=*/(short)0, c8, /*reuse_a=*/false, /*reuse_b=*/false);
    }
    // D layout: lane 0-15: (M=v, N=lane); lane 16-31: (M=v+8, N=lane-16)
    if (nidx < OT_) {
#pragma unroll
      for (int v = 0; v < 8; ++v) {
        const int m    = v + ((lane >> 4) << 3);
        const int cidx = mt * 16 + m;
        float y = c8[v] + bv;
        const float s = 1.0f / (1.0f + __expf(-BETA_ * y));   // Swish
        y *= s;
        out[(((size_t)b * T_OUT_ + (T_IN_ + nidx)) * N_ + n) * C_ + cidx] = y;
      }
    }
  }
}

// ---------------------------------------------------------------------------
extern "C" void kernel_launch(void* const* d_in, const int* in_sizes, int n_in,
                              void* d_out, int out_size, void* d_ws, size_t ws_size,
                              hipStream_t stream) {
  const float* x    = (const float*)d_in[0];
  const int*   nn   = (const int*)d_in[1];
  const float* nd   = (const float*)d_in[2];
  const float* Wm   = (const float*)d_in[3];
  const float* bias = (const float*)d_in[4];
  float* out = (float*)d_out;
  float* ws  = (float*)d_ws;

  gnn_sigma_kernel<<<1, 256, 0, stream>>>(nd, ws);
  gnn_fused_kernel<<<N_, 128, 0, stream>>>(x, nn, nd, Wm, bias, ws, out);
}